// TopKPool_86904368267566
// MI455X (gfx1250) — compile-verified
//
#include <hip/hip_runtime.h>
#include <stdint.h>

// Problem constants (from reference): x is [32, 512, 4096] fp32, k = 64.
#define N      4096
#define K_TOP  64
#define NT     256            // threads per block (8 waves of 32)
#define EPT    (N / NT)       // 16 elements per thread
#define RPW    4              // rows per workgroup (TDM double-buffer pipeline)

typedef unsigned int uint32x4 __attribute__((ext_vector_type(4)));
typedef int          int32x8  __attribute__((ext_vector_type(8)));
typedef int          int32x4  __attribute__((ext_vector_type(4)));

#if __has_builtin(__builtin_amdgcn_tensor_load_to_lds)
#define HAVE_TDM 1
#endif

// Monotonic float-bits -> uint mapping: larger float => larger key.
__device__ __forceinline__ unsigned int orderKey(unsigned int b) {
    unsigned int m = ((int)b >> 31) | 0x80000000u;
    return b ^ m;
}

// Inclusive scan within a wave32 (no barriers, ds_bpermute/shfl based).
__device__ __forceinline__ unsigned int waveInclScan(unsigned int v, unsigned int lane) {
    #pragma unroll
    for (int d = 1; d < 32; d <<= 1) {
        unsigned int n = __shfl_up(v, d, 32);
        if (lane >= (unsigned int)d) v += n;
    }
    return v;
}

// Inclusive scan across NT=256 threads. 2 barriers total.
__device__ __forceinline__ unsigned int blockInclScan(unsigned int v, int t,
                                                      unsigned int* waveSum) {
    const unsigned int lane = (unsigned int)(t & 31);
    const unsigned int wave = (unsigned int)(t >> 5);
    unsigned int ws = waveInclScan(v, lane);
    if (lane == 31) waveSum[wave] = ws;
    __syncthreads();
    if (t < 32) {
        unsigned int s = (lane < 8) ? waveSum[lane] : 0u;
        #pragma unroll
        for (int d = 1; d < 8; d <<= 1) {
            unsigned int n = __shfl_up(s, d, 32);
            if (lane >= (unsigned int)d) s += n;
        }
        if (lane < 8) waveSum[lane] = s;  // inclusive wave sums
    }
    __syncthreads();
    unsigned int off = (wave == 0) ? 0u : waveSum[wave - 1];
    return ws + off;
}

#ifdef HAVE_TDM
// Issue one TDM row load: 4096 fp32 from global rowPtr -> LDS byte offset ldsOff.
__device__ __forceinline__ void tdmLoadRow(const float* rowPtr, unsigned int ldsOff) {
    unsigned long long ga = (unsigned long long)(uintptr_t)rowPtr;
    // D# group 0: count=1 | lds_addr | global_addr | type=2
    uint32x4 g0;
    g0.x = 1u;                                         // count=1, user mode
    g0.y = ldsOff;                                     // lds_addr (bytes)
    g0.z = (unsigned int)(ga & 0xFFFFFFFFull);         // global_addr[31:0]
    g0.w = (unsigned int)((ga >> 32) & 0x01FFFFFFull)  // global_addr[56:32]
           | (2u << 30);                               // type = 2 ("image")
    // D# group 1: data_size=4B, tensor 4096x1, tile 4096x1, stride 4096
    int32x8 g1;
    g1[0] = (int)(2u << 16);        // workgroup_mask=0, data_size=2 (4 bytes)
    g1[1] = (int)(4096u << 16);     // tensor_dim0[15:0] in [31:16]
    g1[2] = (int)(1u << 16);        // tensor_dim0 hi = 0; tensor_dim1 = 1
    g1[3] = (int)(4096u << 16);     // tensor_dim1 hi = 0; tile_dim0 = 4096
    g1[4] = 1;                      // tile_dim1 = 1, tile_dim2 = 0
    g1[5] = 4096;                   // tensor_dim0_stride[31:0]
    g1[6] = 0;
    g1[7] = 0;
    int32x4 g2 = {0, 0, 0, 0};
    int32x4 g3 = {0, 0, 0, 0};
    int32x8 g4 = {0, 0, 0, 0, 0, 0, 0, 0};
    __builtin_amdgcn_tensor_load_to_lds(g0, g1, g2, g3, g4, 0);
}
#endif

__global__ __launch_bounds__(NT) void topk_pool_kernel(const float* __restrict__ x,
                                                       float* __restrict__ out,
                                                       int nrows) {
    __shared__ float vals[2][N];                // double-buffered rows (32 KB)
    __shared__ unsigned int hist[NT];           // 256-bucket histogram
    __shared__ unsigned int scan[NT];           // suffix-scan exchange buffer
    __shared__ unsigned int waveSum[8];         // cross-wave scan combine
    __shared__ unsigned long long argmaxKey;    // (key<<32) | ~idx
    __shared__ unsigned int sh_digit, sh_rem;

    const int t = threadIdx.x;
    const int rowBase = blockIdx.x * RPW;
    if (rowBase >= nrows) return;

#ifdef HAVE_TDM
    if (t < 32) {  // wave 0 drives the DMA pipeline
        tdmLoadRow(x + (size_t)rowBase * N, (unsigned int)(uintptr_t)&vals[0][0]);
        if (RPW > 1 && rowBase + 1 < nrows)
            tdmLoadRow(x + (size_t)(rowBase + 1) * N,
                       (unsigned int)(uintptr_t)&vals[1][0]);
    }
#endif

    for (int r = 0; r < RPW; ++r) {
        const int row = rowBase + r;
        if (row >= nrows) break;  // uniform across block
        float* vbuf = vals[r & 1];
        const bool nextIssued = (r + 1 < RPW) && (row + 1 < nrows);

#ifdef HAVE_TDM
        if (t < 32) {
            // TENSORcnt completes in order: allow 1 outstanding (the prefetch)
            // iff a load for row r+1 has been issued.
            if (nextIssued) __builtin_amdgcn_s_wait_tensorcnt(1);
            else            __builtin_amdgcn_s_wait_tensorcnt(0);
        }
#else
        {
            const float4* src = (const float4*)(x + (size_t)row * N);
            float4* dst = (float4*)vbuf;
            for (int i = t; i < N / 4; i += NT) dst[i] = src[i];
        }
#endif
        if (t == 0) argmaxKey = 0ull;
        __syncthreads();

        // ---------------- argmax with lowest-index tie-break -----------------
        {
            unsigned long long best = 0ull;
            for (int e = 0; e < EPT; ++e) {
                int i = t + e * NT;  // strided: LDS bank-conflict free
                unsigned int u = orderKey(__float_as_uint(vbuf[i]));
                unsigned long long kk =
                    ((unsigned long long)u << 32) | (unsigned int)(~i);
                best = (kk > best) ? kk : best;
            }
            #pragma unroll
            for (int d = 16; d > 0; d >>= 1) {
                unsigned long long o = __shfl_xor(best, d, 32);
                best = (o > best) ? o : best;
            }
            if ((t & 31) == 0) atomicMax(&argmaxKey, best);
        }
        __syncthreads();
        const unsigned int m = ~(unsigned int)(argmaxKey & 0xFFFFFFFFull);

        // ---------------- radix select: k-th largest key ----------------------
        unsigned int prefix = 0;   // fixed high bits so far (right-aligned)
        unsigned int rem = K_TOP;  // still to take among prefix-matching elements
        for (int pass = 0; pass < 4; ++pass) {
            const int shift = 24 - 8 * pass;
            hist[t] = 0;
            __syncthreads();
            for (int e = 0; e < EPT; ++e) {
                int i = t + e * NT;
                unsigned int u = orderKey(__float_as_uint(vbuf[i]));
                if ((pass == 0) || ((u >> (shift + 8)) == prefix))
                    atomicAdd(&hist[(u >> shift) & 0xFFu], 1u);
            }
            __syncthreads();
            // suffix[d] = #elements with digit >= d  (scan of reversed hist)
            unsigned int incl = blockInclScan(hist[NT - 1 - t], t, waveSum);
            scan[t] = incl;  // scan[r] = suffix[255 - r]
            __syncthreads();
            unsigned int sufD  = scan[NT - 1 - t];                      // suffix[t]
            unsigned int sufD1 = (t == NT - 1) ? 0u : scan[NT - 2 - t]; // suffix[t+1]
            if (sufD >= rem && sufD1 < rem) {  // unique crossing digit
                sh_digit = (unsigned int)t;
                sh_rem   = rem - sufD1;
            }
            __syncthreads();
            prefix = (prefix << 8) | sh_digit;
            rem = sh_rem;
        }
        const unsigned int T = prefix;  // threshold key (k-th largest value)
        const unsigned int need = rem;  // #ties (==T) to keep, lowest rotated idx

        // ---------------- ordered compaction to out[row][0..63] ---------------
        // Thread t owns rotated indices j in [t*EPT, t*EPT + EPT).
        unsigned int gtCnt = 0, eqCnt = 0;
        const int jBase = t * EPT;
        for (int e = 0; e < EPT; ++e) {
            int o = (jBase + e + (int)m) & (N - 1);
            unsigned int u = orderKey(__float_as_uint(vbuf[o]));
            gtCnt += (u > T) ? 1u : 0u;
            eqCnt += (u == T) ? 1u : 0u;
        }
        __syncthreads();  // protect waveSum/scan reuse
        unsigned int packed = (gtCnt << 16) | eqCnt;  // gt<=63, eq<=4096: no carry
        unsigned int incl = blockInclScan(packed, t, waveSum);
        unsigned int excl = incl - packed;
        unsigned int gtPref = excl >> 16;
        unsigned int eqPref = excl & 0xFFFFu;
        float* orow = out + (size_t)row * K_TOP;
        for (int e = 0; e < EPT; ++e) {
            int o = (jBase + e + (int)m) & (N - 1);
            float v = vbuf[o];
            unsigned int u = orderKey(__float_as_uint(v));
            if (u > T) {
                unsigned int eqTaken = (eqPref < need) ? eqPref : need;
                orow[gtPref + eqTaken] = v;
                ++gtPref;
            } else if (u == T) {
                if (eqPref < need) orow[gtPref + eqPref] = v;
                ++eqPref;
            }
        }

        __syncthreads();  // everyone done reading vbuf before DMA overwrites it
#ifdef HAVE_TDM
        if (t < 32) {
            int nr = row + 2;
            if (r + 2 < RPW && nr < nrows)
                tdmLoadRow(x + (size_t)nr * N,
                           (unsigned int)(uintptr_t)&vals[r & 1][0]);
        }
#endif
    }
}

extern "C" void kernel_launch(void* const* d_in, const int* in_sizes, int n_in,
                              void* d_out, int out_size, void* d_ws, size_t ws_size,
                              hipStream_t stream) {
    const float* x = (const float*)d_in[0];
    float* out = (float*)d_out;
    (void)d_ws; (void)ws_size; (void)n_in; (void)out_size;
    int nrows = in_sizes[0] / N;  // 32*512 = 16384
    int blocks = (nrows + RPW - 1) / RPW;
    topk_pool_kernel<<<dim3(blocks), dim3(NT), 0, stream>>>(x, out, nrows);
}